// GaussianMixtureEstimator_37031208026218
// MI455X (gfx1250) — compile-verified
//
#include <hip/hip_runtime.h>
#include <math.h>

typedef __attribute__((ext_vector_type(2))) float v2f;
typedef __attribute__((ext_vector_type(8))) float v8f;

#define LOG_2PI 1.8378770664093453f
#define D_DIM 64
#define K_DIM 64

// ---------------------------------------------------------------------------
// Kernel 1: precompute per-component weight matrices and constants.
//   W1[k,d] = invvar * mu            (coefficient of x)
//   W2[k,d] = -0.5 * invvar          (coefficient of x^2)
//   ck[k]   = -0.5*sum_d(logvar + invvar*mu^2) - 0.5*D*log2pi
//             + logprior[k] - logsumexp(logpriors)
// ---------------------------------------------------------------------------
__global__ __launch_bounds__(64) void gmm_precompute(
    const float* __restrict__ mu, const float* __restrict__ logvars,
    const float* __restrict__ logpriors,
    float* __restrict__ W1, float* __restrict__ W2, float* __restrict__ ck) {
  __shared__ float red[K_DIM];
  const int k = threadIdx.x;  // 0..63
  float slv = 0.0f, sq = 0.0f;
  for (int d = 0; d < D_DIM; ++d) {
    const float lv = logvars[k * D_DIM + d];
    const float m  = mu[k * D_DIM + d];
    const float iv = expf(-lv);
    W1[k * D_DIM + d] = iv * m;
    W2[k * D_DIM + d] = -0.5f * iv;
    slv += lv;
    sq  += iv * m * m;
  }
  const float lp = logpriors[k];
  red[k] = lp;
  __syncthreads();
  float mx = red[0];
  for (int i = 1; i < K_DIM; ++i) mx = fmaxf(mx, red[i]);
  float s = 0.0f;
  for (int i = 0; i < K_DIM; ++i) s += expf(red[i] - mx);
  const float lsep = mx + logf(s);
  ck[k] = -0.5f * (slv + sq) - 0.5f * (float)D_DIM * LOG_2PI + lp - lsep;
}

// ---------------------------------------------------------------------------
// Kernel 2: fused dual-GEMM (f32 WMMA 16x16x4) + per-sample log-softmax.
// Block = 256 threads = 8 waves; each wave: 16 samples x all 64 components.
// Accumulator tiles: M = mixture component, N = sample index.
// ---------------------------------------------------------------------------
__global__ __launch_bounds__(256) void gmm_main(
    const float* __restrict__ X,   // (N, 64) row-major
    const float* __restrict__ W1,  // (64, 64)
    const float* __restrict__ W2,  // (64, 64)
    const float* __restrict__ ck,  // (64,)
    float* __restrict__ out,       // (64, N) row-major
    int Ntot) {
  __shared__ float sW1[K_DIM * D_DIM];
  __shared__ float sW2[K_DIM * D_DIM];
  __shared__ float sCk[K_DIM];

  const int tid = threadIdx.x;
  // Stage weights into LDS (32 KB total), float4 vectorized.
  {
    const float4* g1 = (const float4*)W1;
    const float4* g2 = (const float4*)W2;
    float4* s1 = (float4*)sW1;
    float4* s2 = (float4*)sW2;
    for (int i = tid; i < (K_DIM * D_DIM) / 4; i += 256) {
      s1[i] = g1[i];
      s2[i] = g2[i];
    }
    if (tid < K_DIM) sCk[tid] = ck[tid];
  }
  __syncthreads();

  const int wave = tid >> 5;      // 0..7
  const int lane = tid & 31;      // wave32
  const int r = lane & 15;        // sample slot within tile (N index)
  const int h = lane >> 4;        // lane half
  const int n = blockIdx.x * 128 + wave * 16 + r;  // sample index
  const float* xrow = X + (size_t)n * D_DIM;

  // 4 accumulator tiles (M tiles of 16 components each), zero-initialized.
  v8f acc[4] = {};

  // f32 WMMA fragment layout (16x16x4): per lane, VGPR v holds K = d0 + v + 2*h.
  // Each lane therefore loads the consecutive float pair at column d0 + 2h.
#pragma unroll
  for (int d0 = 0; d0 < D_DIM; d0 += 4) {
    const int c = d0 + 2 * h;
    v2f bx = *(const v2f*)(xrow + c);  // B fragment: x values
    v2f bx2 = bx * bx;                 // B fragment: x^2 values
#pragma unroll
    for (int mt = 0; mt < 4; ++mt) {
      const int m = mt * 16 + r;       // component row
      v2f a1 = *(const v2f*)(sW1 + m * D_DIM + c);
      v2f a2 = *(const v2f*)(sW2 + m * D_DIM + c);
      acc[mt] = __builtin_amdgcn_wmma_f32_16x16x4_f32(
          false, a1, false, bx, (short)0, acc[mt], false, false);
      acc[mt] = __builtin_amdgcn_wmma_f32_16x16x4_f32(
          false, a2, false, bx2, (short)0, acc[mt], false, false);
    }
  }

  // Add per-component constants. C/D layout: VGPR j, lane half h -> M = j + 8h.
#pragma unroll
  for (int mt = 0; mt < 4; ++mt)
#pragma unroll
    for (int j = 0; j < 8; ++j)
      acc[mt][j] += sCk[mt * 16 + 8 * h + j];

  // Log-softmax over all 64 components for this sample.
  // This lane holds 32 of them; lanes r and r^16 share the same sample n.
  float mx = -INFINITY;
#pragma unroll
  for (int mt = 0; mt < 4; ++mt)
#pragma unroll
    for (int j = 0; j < 8; ++j) mx = fmaxf(mx, acc[mt][j]);
  mx = fmaxf(mx, __shfl_xor(mx, 16, 32));

  float s = 0.0f;
#pragma unroll
  for (int mt = 0; mt < 4; ++mt)
#pragma unroll
    for (int j = 0; j < 8; ++j) s += expf(acc[mt][j] - mx);
  s += __shfl_xor(s, 16, 32);
  const float lse = mx + logf(s);

  // Store final log-posteriors: out[k * Ntot + n], coalesced across r lanes.
#pragma unroll
  for (int mt = 0; mt < 4; ++mt)
#pragma unroll
    for (int j = 0; j < 8; ++j) {
      const int k = mt * 16 + 8 * h + j;
      out[(size_t)k * (size_t)Ntot + n] = acc[mt][j] - lse;
    }
}

// ---------------------------------------------------------------------------
extern "C" void kernel_launch(void* const* d_in, const int* in_sizes, int n_in,
                              void* d_out, int out_size, void* d_ws, size_t ws_size,
                              hipStream_t stream) {
  const float* X         = (const float*)d_in[0];  // (N, 64)
  const float* mu        = (const float*)d_in[1];  // (64, 64)
  const float* logvars   = (const float*)d_in[2];  // (64, 64)
  const float* logpriors = (const float*)d_in[3];  // (64,)
  float* out = (float*)d_out;                      // (64, N)

  const int N = in_sizes[0] / D_DIM;

  float* ws = (float*)d_ws;
  float* W1 = ws;                      // 64*64 floats
  float* W2 = ws + K_DIM * D_DIM;      // 64*64 floats
  float* ck = ws + 2 * K_DIM * D_DIM;  // 64 floats

  gmm_precompute<<<1, 64, 0, stream>>>(mu, logvars, logpriors, W1, W2, ck);

  const int samples_per_block = 128;  // 8 waves x 16 samples
  gmm_main<<<N / samples_per_block, 256, 0, stream>>>(X, W1, W2, ck, out, N);
}